// DenoisingGNN_7653631721973
// MI455X (gfx1250) — compile-verified
//
#include <hip/hip_runtime.h>
#include <math.h>

#define N_NODES 4096
#define M_OBJ   8192
#define NGRAPH  8
#define H       128
#define ND      16
#define NE      131072

typedef float v8f __attribute__((ext_vector_type(8)));
typedef float v2f __attribute__((ext_vector_type(2)));

// ---- workspace layout (float offsets) ----
#define OFF_GADD   0                       // [8,128] per-graph additive term
#define OFF_CSUM   1024                    // [8,16]  masked sum of cond_x
#define OFF_CCNT   1152                    // [8]     masked counts (float)
#define OFF_NEMB   1280                    // [4096,128] node_emb / reused as h2
#define OFF_XW     (OFF_NEMB + 524288)     // [4096,128] X@W scratch
#define OFF_HBUF   (OFF_XW   + 524288)     // [4096,128] conv accumulator / h1
#define OFF_DEG    (OFF_HBUF + 524288)     // [4096]
#define OFF_DINV   (OFF_DEG  + 4096)       // [4096]
#define OFF_SL     (OFF_DINV + 4096)       // [4096]
#define OFF_SR     (OFF_SL   + 4096)       // [4096]

__global__ void k_init(float* ws) {
    int t = blockIdx.x * blockDim.x + threadIdx.x;
    if (t < 128) ws[OFF_CSUM + t] = 0.0f;
    if (t < 8)   ws[OFF_CCNT + t] = 0.0f;
    if (t < N_NODES) ws[OFF_DEG + t] = 1.0f;
}

__global__ void k_cond_pool(const int* __restrict__ nuc,
                            const int* __restrict__ central,
                            const int* __restrict__ backbone,
                            const float* __restrict__ obj_x,
                            const float* __restrict__ obj_pos,
                            const int* __restrict__ obj_batch,
                            float* __restrict__ csum, float* __restrict__ ccnt) {
    int m = blockIdx.x * blockDim.x + threadIdx.x;
    if (m >= M_OBJ) return;
    bool is_left    = (nuc[m] == 0);
    bool is_central = (central[m] != 0);
    bool is_right   = (nuc[m] == 2);
    bool is_base    = (backbone[m] == 0);
    bool is_cond = is_left | (is_central & is_base) | (is_right & is_base);
    if (!is_cond) return;
    int g = obj_batch[m];
    atomicAdd(&ccnt[g], 1.0f);
    #pragma unroll
    for (int d = 0; d < ND; ++d) {
        float v = (d < 13) ? obj_x[m * 13 + d] : obj_pos[m * 3 + (d - 13)];
        atomicAdd(&csum[g * ND + d], v);
    }
}

__global__ void k_graph_embed(const int* __restrict__ timestep,
                              const float* __restrict__ csum,
                              const float* __restrict__ ccnt,
                              const float* __restrict__ W_node, const float* __restrict__ b_node,
                              const float* __restrict__ W_cond, const float* __restrict__ b_cond,
                              const float* __restrict__ W_time, const float* __restrict__ b_time,
                              float* __restrict__ gadd) {
    __shared__ float te[H];
    __shared__ float pool[H];
    int g = blockIdx.x;
    int h = threadIdx.x;
    float t = (float)timestep[g];
    int j = (h < 64) ? h : (h - 64);
    float f = __expf(-logf(10000.0f) * (float)j / 64.0f);
    float a = t * f;
    te[h] = (h < 64) ? __cosf(a) : __sinf(a);
    float cnt = ccnt[g];
    float mx  = fmaxf(cnt, 1.0f);
    float s = 0.0f;
    #pragma unroll
    for (int d = 0; d < ND; ++d) s += csum[g * ND + d] * W_node[d * H + h];
    s /= mx;
    if (cnt > 0.0f) s += b_node[h];
    pool[h] = s;
    __syncthreads();
    float th = b_time[h];
    float ch = b_cond[h];
    for (int k = 0; k < H; ++k) {
        th += te[k]   * W_time[k * H + h];
        ch += pool[k] * W_cond[k * H + h];
    }
    gadd[g * H + h] = th + ch;
}

__global__ void k_node_embed(const float* __restrict__ x,
                             const int* __restrict__ batch_map,
                             const float* __restrict__ W_node, const float* __restrict__ b_node,
                             const float* __restrict__ gadd,
                             float* __restrict__ nemb) {
    __shared__ float xr[ND];
    __shared__ int bm;
    int i = blockIdx.x;
    int h = threadIdx.x;
    if (h < ND) xr[h] = x[i * ND + h];
    if (h == 0) bm = batch_map[i];
    __syncthreads();
    float v = b_node[h] + gadd[bm * H + h];
    #pragma unroll
    for (int d = 0; d < ND; ++d) v += xr[d] * W_node[d * H + h];
    nemb[i * H + h] = v;
}

__global__ void k_degree(const int* __restrict__ dst, float* __restrict__ deg) {
    int e = blockIdx.x * blockDim.x + threadIdx.x;
    if (e < NE) atomicAdd(&deg[dst[e]], 1.0f);
}
__global__ void k_dinv(const float* __restrict__ deg, float* __restrict__ dinv) {
    int i = blockIdx.x * blockDim.x + threadIdx.x;
    if (i < N_NODES) {
        float d = deg[i];
        dinv[i] = (d > 0.0f) ? rsqrtf(fmaxf(d, 1.0f)) : 0.0f;
    }
}

// C[4096x128] = A[4096x128] @ B[128x128] via V_WMMA_F32_16X16X4_F32.
// One wave per 16x16 output tile, 32 chained WMMAs over K=128.
__global__ void k_gemm128_wmma(const float* __restrict__ A,
                               const float* __restrict__ B,
                               float* __restrict__ C) {
    int tid  = threadIdx.x;
    int wave = tid >> 5;
    int lane = tid & 31;
    int tile = blockIdx.x * (blockDim.x >> 5) + wave;   // 2048 tiles
    int tm = tile >> 3;
    int tn = tile & 7;
    int r    = lane & 15;
    int half = lane >> 4;
    v8f acc = {};
    const float* Arow = A + (tm * 16 + r) * H;
    const float* Bcol = B + tn * 16 + r;
    #pragma unroll 4
    for (int k0 = 0; k0 < H; k0 += 4) {
        int ka = k0 + 2 * half;
        v2f a, b;
        a.x = Arow[ka];
        a.y = Arow[ka + 1];
        b.x = Bcol[ka * H];
        b.y = Bcol[(ka + 1) * H];
        acc = __builtin_amdgcn_wmma_f32_16x16x4_f32(
                false, a, false, b, (short)0, acc, false, false);
    }
    float* Crow = C + (tm * 16 + 8 * half) * H + tn * 16 + r;
    #pragma unroll
    for (int v = 0; v < 8; ++v)
        Crow[v * H] = acc[v];
}

__global__ void k_conv_init(const float* __restrict__ xw,
                            const float* __restrict__ dinv,
                            const float* __restrict__ bias,
                            float* __restrict__ out) {
    int t = blockIdx.x * blockDim.x + threadIdx.x;
    if (t >= N_NODES * H) return;
    int i = t >> 7;
    int c = t & (H - 1);
    float di = dinv[i];
    out[t] = bias[c] + xw[t] * di * di;
}

__global__ void k_conv_scatter(const float* __restrict__ xw,
                               const int* __restrict__ src,
                               const int* __restrict__ dst,
                               const float* __restrict__ dinv,
                               float* __restrict__ out) {
    long long t = (long long)blockIdx.x * blockDim.x + threadIdx.x;
    int e = (int)(t >> 7);
    int c = (int)(t & (H - 1));
    if (e >= NE) return;
    int s = src[e], d = dst[e];
    float nm = dinv[s] * dinv[d];
    atomicAdd(&out[d * H + c], xw[s * H + c] * nm);
}

__global__ void k_relu(float* __restrict__ p, int n) {
    int t = blockIdx.x * blockDim.x + threadIdx.x;
    if (t < n) p[t] = fmaxf(p[t], 0.0f);
}

__global__ void k_out_proj(const float* __restrict__ h2,
                           const float* __restrict__ W_out,
                           const float* __restrict__ b_out,
                           float* __restrict__ out) {
    int t = blockIdx.x * blockDim.x + threadIdx.x;
    if (t >= N_NODES * ND) return;
    int i = t >> 4;
    int d = t & (ND - 1);
    float acc = b_out[d];
    for (int c = 0; c < H; ++c) acc += h2[i * H + c] * W_out[c * ND + d];
    out[t] = acc;
}

__global__ void k_edge_scores(const float* __restrict__ h2,
                              const float* __restrict__ w_edge,
                              float* __restrict__ sl, float* __restrict__ sr) {
    int t = blockIdx.x * blockDim.x + threadIdx.x;
    if (t >= 2 * N_NODES) return;
    int i = t & (N_NODES - 1);
    const float* w = (t < N_NODES) ? w_edge : (w_edge + H);
    float acc = 0.0f;
    for (int c = 0; c < H; ++c) acc += h2[i * H + c] * w[c];
    if (t < N_NODES) sl[i] = acc; else sr[i] = acc;
}

__global__ void k_edge_logits(const float* __restrict__ sl,
                              const float* __restrict__ sr,
                              const float* __restrict__ b_edge,
                              float* __restrict__ out) {
    int i = blockIdx.x;
    long long off = (long long)i * (N_NODES - 1) - (long long)i * (i - 1) / 2;
    float base = sl[i] + b_edge[0];
    for (int j = i + 1 + threadIdx.x; j < N_NODES; j += blockDim.x)
        out[off + (j - i - 1)] = base + sr[j];
}

extern "C" void kernel_launch(void* const* d_in, const int* in_sizes, int n_in,
                              void* d_out, int out_size, void* d_ws, size_t ws_size,
                              hipStream_t stream) {
    const float* x          = (const float*)d_in[0];
    const int*   edge_index = (const int*)  d_in[1];
    const int*   timestep   = (const int*)  d_in[2];
    const int*   batch_map  = (const int*)  d_in[3];
    const int*   nuc        = (const int*)  d_in[4];
    const int*   central    = (const int*)  d_in[5];
    const int*   backbone   = (const int*)  d_in[6];
    const float* obj_x      = (const float*)d_in[7];
    const float* obj_pos    = (const float*)d_in[8];
    const int*   obj_batch  = (const int*)  d_in[9];
    const float* W_node = (const float*)d_in[10]; const float* b_node = (const float*)d_in[11];
    const float* W_cond = (const float*)d_in[12]; const float* b_cond = (const float*)d_in[13];
    const float* W_time = (const float*)d_in[14]; const float* b_time = (const float*)d_in[15];
    const float* W_conv1= (const float*)d_in[16]; const float* b_conv1= (const float*)d_in[17];
    const float* W_conv2= (const float*)d_in[18]; const float* b_conv2= (const float*)d_in[19];
    const float* W_out  = (const float*)d_in[20]; const float* b_out  = (const float*)d_in[21];
    const float* w_edge = (const float*)d_in[22]; const float* b_edge = (const float*)d_in[23];

    float* ws   = (float*)d_ws;
    float* gadd = ws + OFF_GADD;
    float* csum = ws + OFF_CSUM;
    float* ccnt = ws + OFF_CCNT;
    float* nemb = ws + OFF_NEMB;
    float* xw   = ws + OFF_XW;
    float* hbuf = ws + OFF_HBUF;
    float* deg  = ws + OFF_DEG;
    float* dinv = ws + OFF_DINV;
    float* sl   = ws + OFF_SL;
    float* sr   = ws + OFF_SR;

    const int* e_src = edge_index;
    const int* e_dst = edge_index + NE;

    float* out_node = (float*)d_out;
    float* out_elog = out_node + N_NODES * ND;

    k_init<<<16, 256, 0, stream>>>(ws);
    k_cond_pool<<<M_OBJ / 256, 256, 0, stream>>>(nuc, central, backbone, obj_x,
                                                 obj_pos, obj_batch, csum, ccnt);
    k_graph_embed<<<NGRAPH, H, 0, stream>>>(timestep, csum, ccnt,
                                            W_node, b_node, W_cond, b_cond,
                                            W_time, b_time, gadd);
    k_node_embed<<<N_NODES, H, 0, stream>>>(x, batch_map, W_node, b_node, gadd, nemb);
    k_degree<<<NE / 256, 256, 0, stream>>>(e_dst, deg);
    k_dinv<<<N_NODES / 256, 256, 0, stream>>>(deg, dinv);

    // conv1
    k_gemm128_wmma<<<512, 128, 0, stream>>>(nemb, W_conv1, xw);
    k_conv_init<<<(N_NODES * H) / 256, 256, 0, stream>>>(xw, dinv, b_conv1, hbuf);
    k_conv_scatter<<<(NE * H) / 256, 256, 0, stream>>>(xw, e_src, e_dst, dinv, hbuf);
    k_relu<<<(N_NODES * H) / 256, 256, 0, stream>>>(hbuf, N_NODES * H);

    // conv2 (h2 accumulates into reused nemb buffer)
    k_gemm128_wmma<<<512, 128, 0, stream>>>(hbuf, W_conv2, xw);
    k_conv_init<<<(N_NODES * H) / 256, 256, 0, stream>>>(xw, dinv, b_conv2, nemb);
    k_conv_scatter<<<(NE * H) / 256, 256, 0, stream>>>(xw, e_src, e_dst, dinv, nemb);
    k_relu<<<(N_NODES * H) / 256, 256, 0, stream>>>(nemb, N_NODES * H);

    // heads
    k_out_proj<<<(N_NODES * ND) / 256, 256, 0, stream>>>(nemb, W_out, b_out, out_node);
    k_edge_scores<<<(2 * N_NODES) / 256, 256, 0, stream>>>(nemb, w_edge, sl, sr);
    k_edge_logits<<<N_NODES, 256, 0, stream>>>(sl, sr, b_edge, out_elog);
}